// SimpleRNN_13305808683617
// MI455X (gfx1250) — compile-verified
//
#include <hip/hip_runtime.h>
#include <hip/hip_bf16.h>

// Elman RNN on MI455X (gfx1250), fp32 WMMA 16x16x4 for the 8x8 recurrence.
//
// Roofline: 4.3 GFLOP total vs 134 MB of x traffic -> compute/latency bound on
// the T=1024 serial chain. One wave32 owns 16 batch rows; state h^T lives in
// 8 VGPRs for the whole sequence. A-rows 8..15 duplicate W_hh rows 0..7 so the
// D tile's lane halves carry identical data, letting the D->B feedback be pure
// v_cndmask (no cross-lane traffic). Two v_wmma_f32_16x16x4_f32 per step cover
// K=8; tanh uses the gfx1250 transcendental unit (co-executes with WMMA).

typedef float v2f __attribute__((ext_vector_type(2)));
typedef float v8f __attribute__((ext_vector_type(8)));

#define B_SZ 32768
#define T_SZ 1024

#if __has_builtin(__builtin_amdgcn_tanhf)
__device__ __forceinline__ float fast_tanh(float x) { return __builtin_amdgcn_tanhf(x); }
#else
__device__ __forceinline__ float fast_tanh(float x) { return tanhf(x); }
#endif

__global__ __launch_bounds__(256) void rnn_wmma_kernel(
    const float* __restrict__ x,     // [B, T]
    const float* __restrict__ W_ih,  // [8]
    const float* __restrict__ W_hh,  // [8,8]
    const float* __restrict__ b_ih,  // [8]
    const float* __restrict__ b_hh,  // [8]
    const float* __restrict__ fc_w,  // [8]
    const float* __restrict__ fc_b,  // [1]
    float* __restrict__ out)         // [B]
{
    const int lane = threadIdx.x & 31;
    const int wave = (int)((blockIdx.x * blockDim.x + threadIdx.x) >> 5);
    const int b0   = wave * 16;          // 16 batch rows per wave (N dim)
    const int m    = lane & 15;
    const int j    = m & 7;              // W_hh row, duplicated across M halves
    const int half = lane >> 4;
    const bool lo  = (half == 0);
    const int b    = b0 + m;             // batch element mirrored by this lane

    // A operands for D = A x B: A[m][k] = W_hh[m&7][k]; 16x4 f32 layout:
    // V0 holds K= (half?2:0), V1 holds K=(half?3:1); second WMMA covers K=4..7.
    v2f a1, a2;
    a1.x = W_hh[j * 8 + (half ? 2 : 0)];
    a1.y = W_hh[j * 8 + (half ? 3 : 1)];
    a2.x = W_hh[j * 8 + (half ? 6 : 4)];
    a2.y = W_hh[j * 8 + (half ? 7 : 5)];

    float wih[8], bias[8], fcw[8];
#pragma unroll
    for (int r = 0; r < 8; ++r) {
        wih[r]  = W_ih[r];
        bias[r] = b_ih[r] + b_hh[r];
        fcw[r]  = fc_w[r];
    }
    const float fcb = fc_b[0];

    v8f h = {};  // h^T state: h[r] (all lanes) = h[b0+(lane&15)][r]

    // x stream for this lane's batch row, 4 timesteps per 16B load.
    const float4* __restrict__ xv4 =
        reinterpret_cast<const float4*>(x + (size_t)b * T_SZ);
    float4 nxt = xv4[0];

    for (int t4 = 0; t4 < T_SZ / 4; ++t4) {
        const float4 cur = nxt;
        if (t4 + 1 < T_SZ / 4) nxt = xv4[t4 + 1];  // uniform-branch prefetch
        const float xs[4] = {cur.x, cur.y, cur.z, cur.w};

#pragma unroll
        for (int u = 0; u < 4; ++u) {
            const float xv = xs[u];

            // C = x*W_ih + (b_ih+b_hh), rows duplicated across M halves.
            v8f c;
#pragma unroll
            for (int r = 0; r < 8; ++r)
                c[r] = __builtin_fmaf(xv, wih[r], bias[r]);

            // B operand (4x16): V0 = rows K0/K2 per lane half, V1 = K1/K3.
            // Lane halves of h are duplicates, so this is pure v_cndmask.
            v2f bb1, bb2;
            bb1.x = lo ? h[0] : h[2];
            bb1.y = lo ? h[1] : h[3];
            bb2.x = lo ? h[4] : h[6];
            bb2.y = lo ? h[5] : h[7];

            v8f acc = __builtin_amdgcn_wmma_f32_16x16x4_f32(
                false, a1, false, bb1, (short)0, c, false, false);
            acc = __builtin_amdgcn_wmma_f32_16x16x4_f32(
                false, a2, false, bb2, (short)0, acc, false, false);

#pragma unroll
            for (int r = 0; r < 8; ++r)
                h[r] = fast_tanh(acc[r]);
        }
    }

    // Head: out[b] = sum_j h[b][j]*fc_w[j] + fc_b  (lanes 0..15 only)
    if (lane < 16) {
        float s = fcb;
#pragma unroll
        for (int r = 0; r < 8; ++r)
            s = __builtin_fmaf(h[r], fcw[r], s);
        out[b] = s;
    }
}

extern "C" void kernel_launch(void* const* d_in, const int* in_sizes, int n_in,
                              void* d_out, int out_size, void* d_ws, size_t ws_size,
                              hipStream_t stream) {
    (void)in_sizes; (void)n_in; (void)d_ws; (void)ws_size; (void)out_size;
    const float* x    = (const float*)d_in[0];
    const float* Wih  = (const float*)d_in[1];
    const float* Whh  = (const float*)d_in[2];
    const float* bih  = (const float*)d_in[3];
    const float* bhh  = (const float*)d_in[4];
    const float* fcw  = (const float*)d_in[5];
    const float* fcb  = (const float*)d_in[6];
    float* out = (float*)d_out;

    // 16 batch rows per wave, 8 waves per block -> 2048 waves / 256 blocks.
    dim3 grid(B_SZ / (16 * 8));
    dim3 block(256);
    rnn_wmma_kernel<<<grid, block, 0, stream>>>(x, Wih, Whh, bih, bhh, fcw, fcb, out);
}